// RNNModel_88639535055432
// MI455X (gfx1250) — compile-verified
//
#include <hip/hip_runtime.h>
#include <hip/hip_bf16.h>

// ---------------------------------------------------------------------------
// MI455X (gfx1250) implementation of the RIMs-style RNN reference.
//
// Roofline: decoder GEMM (4096x32000x600 fp32) = 157 GFLOP with a 524 MB
// output -> HBM-bound (~23 us at 23.3 TB/s). fp32 WMMA (V_WMMA_F32_16X16X4_F32)
// is far below the matrix ceiling, so full fp32 precision everywhere.
// Sequential scans: launch-latency bound -> GEMMs fused/batched per step.
// All GEMM inner loops are branch-free (clamped addresses, no exec churn).
// ---------------------------------------------------------------------------

typedef float v2f __attribute__((ext_vector_type(2)));
typedef float v8f __attribute__((ext_vector_type(8)));

#define T_   64
#define B_   64
#define NTOK 32000
#define NHID 600
#define NB_  6
#define BS_  100
#define ATT_ 400

__device__ __forceinline__ float sigf(float x) { return 1.f / (1.f + expf(-x)); }

// ---------------------------------------------------------------------------
// Generic fp32 WMMA GEMM:  C[z] (=?+) A[z] @ B[z] (+ bias[z])
// block = 128 threads = 4 waves, one 16x16 tile per wave.
// Requirements (all satisfied by every call site): K % 4 == 0, lda/ldb even.
// Partial N tiles handled by clamping load addresses (WMMA columns are
// independent, so garbage in never-stored columns is harmless); fully-OOB
// tiles exit early (wave-uniform). No branches in the K loop.
// ---------------------------------------------------------------------------
__global__ void gemm_wmma(const float* __restrict__ A, const float* __restrict__ B,
                          const float* __restrict__ bias, float* __restrict__ C,
                          int M, int N, int K, int lda, int ldb, int ldc,
                          long bsA, long bsB, long bsC, long bsBias,
                          int opB, int accum)
{
    long z = blockIdx.z;
    A += z * bsA; B += z * bsB; C += z * bsC;
    if (bias) bias += z * bsBias;

    int wave = threadIdx.x >> 5;
    int lane = threadIdx.x & 31;
    int hh = lane >> 4;       // half-wave: K pair select / M offset
    int ln = lane & 15;
    int m0 = blockIdx.y * 16;
    int n0 = (blockIdx.x * 4 + wave) * 16;
    if (n0 >= N) return;                       // wave-uniform: whole tile OOB

    int row_a  = m0 + ln; if (row_a > M - 1) row_a = M - 1;   // clamp for loads
    int col    = n0 + ln;
    int col_ld = col > N - 1 ? N - 1 : col;

    const float* Ap = A + (long)row_a * lda + 2 * hh;
    const float* Bp = opB ? (B + (long)col_ld * ldb + 2 * hh)
                          : (B + (long)(2 * hh) * ldb + col_ld);

    v8f acc = {};
    if (accum) {
        #pragma unroll
        for (int r = 0; r < 8; ++r) {
            int row = m0 + r + 8 * hh; if (row > M - 1) row = M - 1;
            acc[r] = C[(long)row * ldc + col_ld];
        }
    }

    if (!opB) {
        for (int kb = 0; kb < K; kb += 4) {
            v2f a = *(const v2f*)(Ap + kb);
            v2f b;
            b[0] = Bp[(long)kb * ldb];
            b[1] = Bp[(long)(kb + 1) * ldb];
            acc = __builtin_amdgcn_wmma_f32_16x16x4_f32(false, a, false, b,
                                                        (short)0, acc, false, false);
        }
    } else {
        for (int kb = 0; kb < K; kb += 4) {
            v2f a = *(const v2f*)(Ap + kb);
            v2f b = *(const v2f*)(Bp + kb);
            acc = __builtin_amdgcn_wmma_f32_16x16x4_f32(false, a, false, b,
                                                        (short)0, acc, false, false);
        }
    }

    #pragma unroll
    for (int r = 0; r < 8; ++r) {
        int row = m0 + r + 8 * hh;
        if (row < M && col < N) {
            float v = acc[r];
            if (bias) v += bias[col];
            C[(long)row * ldc + col] = v;
        }
    }
}

// ---------------------------------------------------------------------------
// Tiled fp32 WMMA GEMM for the two big GEMMs (X@Wih0 and decoder).
// 256 threads = 8 waves; macro-tile M=64 x N=128; K chunks of 8 staged in LDS;
// each wave register-blocks a 16x64 strip (4 x v8f accumulators).
// Requires: M % 64 == 0, K % 8 == 0, lda=K, ldb=ldc=N. N edge: clamped loads,
// guarded stores. Prefetches the next B chunk (global_prefetch_b8).
// ---------------------------------------------------------------------------
__global__ __launch_bounds__(256)
void gemm_wmma_tiled(const float* __restrict__ A, const float* __restrict__ B,
                     const float* __restrict__ bias, float* __restrict__ C,
                     int M, int N, int K)
{
    __shared__ float As[8][64];
    __shared__ float Bs[8][128];

    int tid  = threadIdx.x;
    int wave = tid >> 5, lane = tid & 31;
    int hh = lane >> 4, ln = lane & 15;
    int m0 = blockIdx.y * 64;
    int n0 = blockIdx.x * 128;
    int mt = wave & 3;     // which 16-row M tile
    int nh = wave >> 2;    // which 64-col half

    v8f acc[4] = {};

    for (int kb = 0; kb < K; kb += 8) {
        __syncthreads();
        {   // stage A chunk (64x8), transposed into LDS
            int e = tid;
            #pragma unroll
            for (int i = 0; i < 2; ++i, e += 256) {
                int m = e >> 3, kk = e & 7;
                As[kk][m] = A[(long)(m0 + m) * K + kb + kk];
            }
        }
        {   // stage B chunk (8x128); clamp N edge (never-stored cols)
            int e = tid;
            #pragma unroll
            for (int i = 0; i < 4; ++i, e += 256) {
                int kk = e >> 7, n = e & 127;
                int gn = n0 + n; if (gn > N - 1) gn = N - 1;
                Bs[kk][n] = B[(long)(kb + kk) * N + gn];
            }
        }
        if (kb + 8 < K) {   // prefetch next B chunk into cache
            int gn = n0 + (tid & 127); if (gn > N - 1) gn = N - 1;
            __builtin_prefetch(B + (long)(kb + 8 + (tid >> 7) * 4) * N + gn, 0, 1);
        }
        __syncthreads();

        #pragma unroll
        for (int s = 0; s < 2; ++s) {
            int kl = 4 * s + 2 * hh;
            v2f a;
            a[0] = As[kl][mt * 16 + ln];
            a[1] = As[kl + 1][mt * 16 + ln];
            #pragma unroll
            for (int j = 0; j < 4; ++j) {
                int nc = nh * 64 + j * 16 + ln;
                v2f b;
                b[0] = Bs[kl][nc];
                b[1] = Bs[kl + 1][nc];
                acc[j] = __builtin_amdgcn_wmma_f32_16x16x4_f32(false, a, false, b,
                                                               (short)0, acc[j], false, false);
            }
        }
    }

    #pragma unroll
    for (int j = 0; j < 4; ++j)
        #pragma unroll
        for (int r = 0; r < 8; ++r) {
            int row = m0 + mt * 16 + r + 8 * hh;
            int col = n0 + nh * 64 + j * 16 + ln;
            if (col < N) C[(long)row * N + col] = acc[j][r] + bias[col];
        }
}

// ---------------------------------------------------------------------------
// Small kernels
// ---------------------------------------------------------------------------
__global__ void k_gather(const int* __restrict__ inp, const float* __restrict__ emb,
                         float* __restrict__ X)
{
    int r = blockIdx.x;
    long tok = inp[r];
    for (int j = threadIdx.x; j < NHID; j += blockDim.x)
        X[(long)r * NHID + j] = emb[tok * NHID + j];
}

// column-packing: dst[r, col_off + c] = src[r, c]
__global__ void k_pack_cols(const float* __restrict__ src, float* __restrict__ dst,
                            int rows, int cols, int ld_dst, int col_off)
{
    long id = blockIdx.x * (long)blockDim.x + threadIdx.x;
    if (id >= (long)rows * cols) return;
    int r = id / cols, cidx = id % cols;
    dst[(long)r * ld_dst + col_off + cidx] = src[id];
}

__global__ void k_lstm0_gate(const float* __restrict__ G, float* __restrict__ h,
                             float* __restrict__ c, float* __restrict__ hs0t)
{
    int id = blockIdx.x * blockDim.x + threadIdx.x;
    if (id >= B_ * NHID) return;
    int b = id / NHID, j = id % NHID;
    const float* g = G + (long)b * 4 * NHID;
    float gi = g[j], gf = g[NHID + j], gg = g[2 * NHID + j], go = g[3 * NHID + j];
    float cn = sigf(gf) * c[id] + sigf(gi) * tanhf(gg);
    float hn = sigf(go) * tanhf(cn);
    c[id] = cn; h[id] = hn; hs0t[id] = hn;
}

// Fused input attention (2 kv slots; slot1 = null -> k=bk1, v=bv1), top-k
// null-routing mask, and value mixing. grid=B, block=192 (6 wave32 warps).
// kv: [B,464] = [k0 (64) | v0 (400)] from the packed wk1|wv1 GEMM.
__global__ void k_attn_av(const float* __restrict__ q, const float* __restrict__ kv,
                          const float* __restrict__ bk1, const float* __restrict__ bv1,
                          float* __restrict__ av, float* __restrict__ maskb)
{
    __shared__ float a0s[NB_], a1s[NB_];
    int b = blockIdx.x;
    int w = threadIdx.x >> 5, lane = threadIdx.x & 31;
    int r = b * NB_ + w;
    float q0 = q[r * 64 + lane], q1 = q[r * 64 + lane + 32];
    float s0 = q0 * kv[b * 464 + lane] + q1 * kv[b * 464 + lane + 32];
    float s1 = q0 * bk1[lane] + q1 * bk1[lane + 32];
    for (int off = 16; off; off >>= 1) {
        s0 += __shfl_xor(s0, off);
        s1 += __shfl_xor(s1, off);
    }
    if (lane == 0) {
        s0 *= 0.125f; s1 *= 0.125f;                 // 1/sqrt(64)
        float m = fmaxf(s0, s1);
        float e0 = expf(s0 - m), e1 = expf(s1 - m);
        float inv = 1.f / (e0 + e1);
        a0s[w] = e0 * inv; a1s[w] = e1 * inv;
    }
    __syncthreads();
    if (threadIdx.x < NB_) {
        int k = threadIdx.x;
        float ak = a1s[k];
        int cnt = 0;
        for (int j = 0; j < NB_; ++j)
            cnt += (a1s[j] > ak) || (a1s[j] == ak && j < k);   // stable top_k order
        maskb[b * NB_ + k] = (cnt < (NB_ - 4)) ? 0.f : 1.f;    // drop top-2 null-attn
    }
    __syncthreads();
    for (int e = threadIdx.x; e < NB_ * ATT_; e += blockDim.x) {
        int k = e / ATT_, i = e % ATT_;
        av[(long)(b * NB_ + k) * ATT_ + i] = a0s[k] * kv[b * 464 + 64 + i] + a1s[k] * bv1[i];
    }
}

__global__ void k_blk_gate(const float* __restrict__ gbuf, const float* __restrict__ c,
                           float* __restrict__ hn, float* __restrict__ cn)
{
    int id = blockIdx.x * blockDim.x + threadIdx.x;
    if (id >= B_ * NHID) return;
    int b = id / NHID, rem = id % NHID, k = rem / BS_, j = rem % BS_;
    const float* g = gbuf + (long)(b * NB_ + k) * 4 * BS_;
    float cold = c[id];
    float cnew = sigf(g[BS_ + j]) * cold + sigf(g[j]) * tanhf(g[2 * BS_ + j]);
    float hnew = sigf(g[3 * BS_ + j]) * tanhf(cnew);
    hn[id] = hnew; cn[id] = cnew;
}

// communication attention: 4 heads x 16 dims over 6 blocks; 1 thread per (b,q,h)
__global__ void k_comm_attn(const float* __restrict__ q2, const float* __restrict__ k2,
                            const float* __restrict__ v2, float* __restrict__ o2)
{
    int id = blockIdx.x * blockDim.x + threadIdx.x;
    if (id >= B_ * NB_ * 4) return;
    int b = id / 24, rem = id % 24, qk = rem / 4, h = rem % 4;
    const float* qp = q2 + (long)(b * NB_ + qk) * 64 + h * 16;
    float sc[NB_], m = -1e30f;
    for (int kk = 0; kk < NB_; ++kk) {
        const float* kp = k2 + (long)(b * NB_ + kk) * 64 + h * 16;
        float s = 0.f;
        for (int d = 0; d < 16; ++d) s += qp[d] * kp[d];
        s *= 0.25f;                                    // 1/sqrt(16)
        sc[kk] = s; m = fmaxf(m, s);
    }
    float sum = 0.f;
    for (int kk = 0; kk < NB_; ++kk) { sc[kk] = expf(sc[kk] - m); sum += sc[kk]; }
    float inv = 1.f / sum;
    float* op = o2 + (long)(b * NB_ + qk) * 64 + h * 16;
    for (int d = 0; d < 16; ++d) {
        float acc = 0.f;
        for (int kk = 0; kk < NB_; ++kk)
            acc += sc[kk] * v2[(long)(b * NB_ + kk) * 64 + h * 16 + d];
        op[d] = acc * inv;
    }
}

__global__ void k_final(const float* __restrict__ gg2, const float* __restrict__ ff2,
                        const float* __restrict__ hn, const float* __restrict__ cn,
                        const float* __restrict__ maskb,
                        float* __restrict__ h, float* __restrict__ c,
                        float* __restrict__ hs1t)
{
    int id = blockIdx.x * blockDim.x + threadIdx.x;
    if (id >= B_ * NHID) return;
    int b = id / NHID, rem = id % NHID, k = rem / BS_, j = rem % BS_;
    int r = b * NB_ + k;
    float comm = sigf(gg2[r * BS_ + j]) * tanhf(ff2[r * BS_ + j]);
    float hfull = hn[id] + comm;
    float m = maskb[r];
    float ho = m * hfull + (1.f - m) * h[id];
    float co = m * cn[id] + (1.f - m) * c[id];
    h[id] = ho; c[id] = co; hs1t[id] = ho;
}

// ---------------------------------------------------------------------------
// Host side
// ---------------------------------------------------------------------------
static inline void launch_gemm(hipStream_t s, const float* A, const float* B,
                               const float* bias, float* C,
                               int M, int N, int K, int lda, int ldb, int ldc,
                               long bsA = 0, long bsB = 0, long bsC = 0, long bsBias = 0,
                               int batch = 1, int opB = 0, int accum = 0)
{
    dim3 g((N + 63) / 64, (M + 15) / 16, batch);
    gemm_wmma<<<g, 128, 0, s>>>(A, B, bias, C, M, N, K, lda, ldb, ldc,
                                bsA, bsB, bsC, bsBias, opB, accum);
}

extern "C" void kernel_launch(void* const* d_in, const int* in_sizes, int n_in,
                              void* d_out, int out_size, void* d_ws, size_t ws_size,
                              hipStream_t stream)
{
    const int*   input = (const int*)  d_in[0];
    const float* h0    = (const float*)d_in[1];
    const float* c0    = (const float*)d_in[2];
    const float* emb   = (const float*)d_in[3];
    const float* Wih0  = (const float*)d_in[4];
    const float* Whh0  = (const float*)d_in[5];
    const float* b0    = (const float*)d_in[6];
    const float* wq1   = (const float*)d_in[7];
    const float* bq1   = (const float*)d_in[8];
    const float* wk1   = (const float*)d_in[9];
    const float* bk1   = (const float*)d_in[10];
    const float* wv1   = (const float*)d_in[11];
    const float* bv1   = (const float*)d_in[12];
    const float* wfc1  = (const float*)d_in[13];
    const float* bfc1  = (const float*)d_in[14];
    const float* Wihb  = (const float*)d_in[15];
    const float* Whhb  = (const float*)d_in[16];
    const float* bb    = (const float*)d_in[17];
    const float* wq2   = (const float*)d_in[18];
    const float* bq2   = (const float*)d_in[19];
    const float* wk2   = (const float*)d_in[20];
    const float* bk2   = (const float*)d_in[21];
    const float* wv2   = (const float*)d_in[22];
    const float* bv2   = (const float*)d_in[23];
    const float* wfc2  = (const float*)d_in[24];
    const float* bfc2  = (const float*)d_in[25];
    const float* wg2   = (const float*)d_in[26];
    const float* bg2   = (const float*)d_in[27];
    const float* Wdec  = (const float*)d_in[28];
    const float* bdec  = (const float*)d_in[29];
    float* out = (float*)d_out;

    const int TB = T_ * B_;                 // 4096
    const size_t ST = (size_t)B_ * NHID;    // 38400 per-layer state

    float* p     = (float*)d_ws;
    float* X     = p; p += (size_t)TB * NHID;         // embedded input
    float* XW    = p; p += (size_t)TB * 4 * NHID;     // x@Wih0+b0 (then += h@Whh0)
    float* hs0   = p; p += (size_t)TB * NHID;         // layer-0 outputs
    float* hs1   = p; p += (size_t)TB * NHID;         // layer-1 outputs
    float* h0s   = p; p += ST;
    float* c0s   = p; p += ST;
    float* h1s   = p; p += ST;
    float* c1s   = p; p += ST;
    float* q1    = p; p += 384 * 64;
    float* kvb   = p; p += 64 * 464;                  // [k0(64)|v0(400)]
    float* maskb = p; p += 384;
    float* avb   = p; p += 384 * ATT_;
    float* iub   = p; p += 384 * ATT_;
    float* gb    = p; p += 384 * 400;
    float* hnb   = p; p += ST;
    float* cnb   = p; p += ST;
    float* qkv2b = p; p += 3 * 384 * 64;              // [q2|k2|v2]
    float* o2b   = p; p += 384 * 64;
    float* gf2b  = p; p += 2 * 384 * BS_;             // [gg2|ff2]
    // packed weights (constant across steps, packed once per call)
    float* wkv1  = p; p += (size_t)NHID * 464;        // [wk1|wv1]
    float* bkv1  = p; p += 464;
    float* wqkv2 = p; p += 3 * BS_ * 64;
    float* bqkv2 = p; p += 3 * 64;
    float* wgf2  = p; p += 2 * 64 * BS_;
    float* bgf2  = p; p += 2 * BS_;

    // init recurrent states from h0/c0 inputs ([2,B,NHID])
    hipMemcpyAsync(h0s, h0,      ST * sizeof(float), hipMemcpyDeviceToDevice, stream);
    hipMemcpyAsync(h1s, h0 + ST, ST * sizeof(float), hipMemcpyDeviceToDevice, stream);
    hipMemcpyAsync(c0s, c0,      ST * sizeof(float), hipMemcpyDeviceToDevice, stream);
    hipMemcpyAsync(c1s, c0 + ST, ST * sizeof(float), hipMemcpyDeviceToDevice, stream);

    // ---- pack step-invariant weights ----
    k_pack_cols<<<(NHID * 64 + 255) / 256, 256, 0, stream>>>(wk1, wkv1, NHID, 64, 464, 0);
    k_pack_cols<<<(NHID * 400 + 255) / 256, 256, 0, stream>>>(wv1, wkv1, NHID, 400, 464, 64);
    hipMemcpyAsync(bkv1,       bk1, 64  * sizeof(float), hipMemcpyDeviceToDevice, stream);
    hipMemcpyAsync(bkv1 + 64,  bv1, 400 * sizeof(float), hipMemcpyDeviceToDevice, stream);
    hipMemcpyAsync(wqkv2,          wq2, 6400 * sizeof(float), hipMemcpyDeviceToDevice, stream);
    hipMemcpyAsync(wqkv2 + 6400,   wk2, 6400 * sizeof(float), hipMemcpyDeviceToDevice, stream);
    hipMemcpyAsync(wqkv2 + 12800,  wv2, 6400 * sizeof(float), hipMemcpyDeviceToDevice, stream);
    hipMemcpyAsync(bqkv2,          bq2, 64 * sizeof(float), hipMemcpyDeviceToDevice, stream);
    hipMemcpyAsync(bqkv2 + 64,     bk2, 64 * sizeof(float), hipMemcpyDeviceToDevice, stream);
    hipMemcpyAsync(bqkv2 + 128,    bv2, 64 * sizeof(float), hipMemcpyDeviceToDevice, stream);
    hipMemcpyAsync(wgf2,           wg2,  6400 * sizeof(float), hipMemcpyDeviceToDevice, stream);
    hipMemcpyAsync(wgf2 + 6400,    wfc2, 6400 * sizeof(float), hipMemcpyDeviceToDevice, stream);
    hipMemcpyAsync(bgf2,           bg2,  BS_ * sizeof(float), hipMemcpyDeviceToDevice, stream);
    hipMemcpyAsync(bgf2 + BS_,     bfc2, BS_ * sizeof(float), hipMemcpyDeviceToDevice, stream);

    // ---- embedding gather + hoisted input-projection GEMM (WMMA tiled) ----
    k_gather<<<TB, 256, 0, stream>>>(input, emb, X);
    gemm_wmma_tiled<<<dim3((4 * NHID + 127) / 128, TB / 64), 256, 0, stream>>>(
        X, Wih0, b0, XW, TB, 4 * NHID, NHID);

    // ---- layer-0 LSTM scan: gates = XW[t] + h @ Whh0 ----
    for (int t = 0; t < T_; ++t) {
        float* G = XW + (size_t)t * B_ * 4 * NHID;
        launch_gemm(stream, h0s, Whh0, nullptr, G, B_, 4 * NHID, NHID,
                    NHID, 4 * NHID, 4 * NHID, 0, 0, 0, 0, 1, 0, /*accum=*/1);
        k_lstm0_gate<<<150, 256, 0, stream>>>(G, h0s, c0s, hs0 + (size_t)t * ST);
    }

    // ---- blocks_step scan ----
    for (int t = 0; t < T_; ++t) {
        const float* xt = hs0 + (size_t)t * ST;
        // q from hidden blocks; packed k|v from input slot
        launch_gemm(stream, h1s, wq1, bq1, q1, 384, 64, BS_, BS_, 64, 64);
        launch_gemm(stream, xt, wkv1, bkv1, kvb, 64, 464, NHID, NHID, 464, 464);
        k_attn_av<<<B_, 192, 0, stream>>>(q1, kvb, bk1, bv1, avb, maskb);
        launch_gemm(stream, avb, wfc1, bfc1, iub, 384, ATT_, ATT_, ATT_, ATT_, ATT_);
        // block-diagonal LSTM gates: g = iu @ Wihb^T  (+  hb @ Whhb^T + bb)
        launch_gemm(stream, iub, Wihb, nullptr, gb, 64, 400, ATT_,
                    NB_ * ATT_, ATT_, NB_ * 400,
                    /*bsA*/ATT_, /*bsB*/(long)400 * ATT_, /*bsC*/400, 0,
                    NB_, /*opB=*/1, /*accum=*/0);
        launch_gemm(stream, h1s, Whhb, bb, gb, 64, 400, BS_,
                    NHID, BS_, NB_ * 400,
                    /*bsA*/BS_, /*bsB*/(long)400 * BS_, /*bsC*/400, /*bsBias*/400,
                    NB_, /*opB=*/1, /*accum=*/1);
        k_blk_gate<<<150, 256, 0, stream>>>(gb, c1s, hnb, cnb);
        // communication attention: batched q2|k2|v2 projection, then attn
        launch_gemm(stream, hnb, wqkv2, bqkv2, qkv2b, 384, 64, BS_, BS_, 64, 64,
                    0, (long)BS_ * 64, 384 * 64, 64, /*batch=*/3);
        k_comm_attn<<<6, 256, 0, stream>>>(qkv2b, qkv2b + 384 * 64, qkv2b + 2 * 384 * 64, o2b);
        launch_gemm(stream, o2b, wgf2, bgf2, gf2b, 384, BS_, 64, 64, BS_, BS_,
                    0, (long)64 * BS_, 384 * BS_, BS_, /*batch=*/2);
        k_final<<<150, 256, 0, stream>>>(gf2b, gf2b + 384 * BS_, hnb, cnb, maskb,
                                         h1s, c1s, hs1 + (size_t)t * ST);
    }

    // ---- decoder: dec = hs1 @ Wdec + bdec  (HBM-bound; WMMA fp32, LDS-tiled)
    gemm_wmma_tiled<<<dim3(NTOK / 128, TB / 64), 256, 0, stream>>>(
        hs1, Wdec, bdec, out, TB, NTOK, NHID);

    // ---- pack final states: [dec | h1,h2 | c1,c2] ----
    const size_t DEC = (size_t)TB * NTOK;   // 131,072,000
    hipMemcpyAsync(out + DEC,          h0s, ST * sizeof(float), hipMemcpyDeviceToDevice, stream);
    hipMemcpyAsync(out + DEC + ST,     h1s, ST * sizeof(float), hipMemcpyDeviceToDevice, stream);
    hipMemcpyAsync(out + DEC + 2 * ST, c0s, ST * sizeof(float), hipMemcpyDeviceToDevice, stream);
    hipMemcpyAsync(out + DEC + 3 * ST, c1s, ST * sizeof(float), hipMemcpyDeviceToDevice, stream);

    (void)in_sizes; (void)n_in; (void)out_size; (void)ws_size;
}